// MVPNet2D_57930518888874
// MI455X (gfx1250) — compile-verified
//
#include <hip/hip_runtime.h>
#include <hip/hip_bf16.h>

typedef __attribute__((ext_vector_type(2))) float v2f;
typedef __attribute__((ext_vector_type(8))) float v8f;

#define NV      5
#define CIN     3
#define IMG_H   120
#define IMG_W   160
#define HW      (IMG_H * IMG_W)     /* 19200  */
#define NVHW    (NV * HW)           /* 96000  */
#define NPOINT  131072
#define KNN     3
#define NCLS    20
#define BATCH   8

// One wave (32 lanes) produces a 20 x 16 output tile (16 points) via two
// V_WMMA_F32_16X16X4_F32 ops (channel rows 0-15 and 16-19 zero padded).
__global__ __launch_bounds__(256) void mvpnet_fused_wmma(
    const float*     __restrict__ images,  // [B, NV, CIN, H, W]
    const long long* __restrict__ knn,     // [B, NPOINT, KNN] int64
    const float*     __restrict__ W2d,     // [NCLS, CIN]
    const float*     __restrict__ b2d,     // [NCLS]
    float*           __restrict__ out)     // [B, NCLS, NPOINT]
{
    const int lane  = threadIdx.x & 31;
    const int wave  = threadIdx.x >> 5;
    const int group = blockIdx.x * 8 + wave;      // 65536 groups of 16 points
    const int groups_per_b = NPOINT / 16;         // 8192
    const int b      = group / groups_per_b;
    const int p_base = (group - b * groups_per_b) * 16;

    const int  n  = lane & 15;                    // point-in-tile / row-in-tile
    const bool hi = lane >= 16;                   // K-half selector (K2,K3)
    const int  p  = p_base + n;

    // ---- A matrices: W2d (20x3), K padded to 4 with zeros -------------------
    // A layout (16x4 f32): lanes 0-15 carry K0(K= vgpr0),K1(vgpr1);
    //                      lanes 16-31 carry K2,K3.
    v2f a0, a1;
    {
        const int m0 = n;                         // rows 0..15
        a0.x = hi ? W2d[m0 * 3 + 2] : W2d[m0 * 3 + 0];
        a0.y = hi ? 0.0f            : W2d[m0 * 3 + 1];
        const int  m1 = 16 + n;                   // rows 16..31 (valid < 20)
        const bool v1 = m1 < NCLS;
        a1.x = v1 ? (hi ? W2d[m1 * 3 + 2] : W2d[m1 * 3 + 0]) : 0.0f;
        a1.y = (v1 && !hi) ? W2d[m1 * 3 + 1] : 0.0f;
    }

    // ---- C accumulators pre-loaded with bias --------------------------------
    // C/D layout: vgpr j, lane l -> M = j + 8*(l>=16), N = l&15
    v8f c0, c1;
#pragma unroll
    for (int j = 0; j < 8; ++j) {
        const int m  = j + (hi ? 8 : 0);
        c0[j] = b2d[m];
        const int m1 = 16 + m;
        c1[j] = (m1 < NCLS) ? b2d[m1] : 0.0f;
    }

    // ---- gather 3 neighbors, average channels -------------------------------
    // flat idx in [0, NVHW): v = idx / HW, pix = idx % HW
    // addr within batch = idx + 2*v*HW + ch*HW   (images are [v][ch][pix])
    const long long* kptr = knn + ((long long)b * NPOINT + p) * KNN;
    const float*     imgb = images + (long long)b * (NV * CIN * HW);

    float sa = 0.0f, sb = 0.0f;
#pragma unroll
    for (int k = 0; k < KNN; ++k) {
        const int idx  = (int)__builtin_nontemporal_load(&kptr[k]);
        const int v    = idx / HW;
        const int base = idx + v * (2 * HW);
        const int choff = hi ? 2 * HW : 0;        // lanes<16: ch0/ch1, lanes>=16: ch2
        sa += imgb[base + choff];
        sb += imgb[base + HW];                    // ch1 (hi lanes: masked below)
    }
    const float inv3 = 1.0f / 3.0f;
    v2f bm;
    bm.x = sa * inv3;                             // B: K0 (lo) / K2 (hi)
    bm.y = hi ? 0.0f : sb * inv3;                 // B: K1 (lo) / K3 pad (hi)

    // ---- 20x16 tile: D = A * B + C  (two 16x16x4 f32 WMMAs) -----------------
    v8f d0 = __builtin_amdgcn_wmma_f32_16x16x4_f32(
        false, a0, false, bm, (short)0, c0, false, false);
    v8f d1 = __builtin_amdgcn_wmma_f32_16x16x4_f32(
        false, a1, false, bm, (short)0, c1, false, false);

    // ---- store: out[b][ch][p], non-temporal (84 MB, don't pollute L2) -------
    float* outb = out + (long long)b * (NCLS * NPOINT) + p;
#pragma unroll
    for (int j = 0; j < 8; ++j) {
        const int ch = j + (hi ? 8 : 0);
        __builtin_nontemporal_store(d0[j], &outb[(long long)ch * NPOINT]);
    }
#pragma unroll
    for (int j = 0; j < 4; ++j) {                 // tile1 rows 16..19 only
        if (!hi)
            __builtin_nontemporal_store(d1[j], &outb[(long long)(16 + j) * NPOINT]);
    }
}

extern "C" void kernel_launch(void* const* d_in, const int* in_sizes, int n_in,
                              void* d_out, int out_size, void* d_ws, size_t ws_size,
                              hipStream_t stream) {
    const float*     images = (const float*)d_in[0];
    const long long* knn    = (const long long*)d_in[1];
    const float*     W2d    = (const float*)d_in[2];
    const float*     b2d    = (const float*)d_in[3];
    float*           out    = (float*)d_out;

    const int total_groups = BATCH * (NPOINT / 16); // 65536 wave-units
    const int blocks       = total_groups / 8;      // 8 waves / 256-thread block
    mvpnet_fused_wmma<<<blocks, 256, 0, stream>>>(images, knn, W2d, b2d, out);
}